// RHF_35708358099212
// MI455X (gfx1250) — compile-verified
//
#include <hip/hip_runtime.h>
#include <hip/hip_bf16.h>

#define NBF    128
#define NBF2   (NBF * NBF)
#define NDOCC_F 16.0f
#define N_ITER 20
#define N_PUR  60   // canonical purification steps (replaces eigh; GEMM-only)
#define CBLK   32   // c-block per jk workgroup (4 blocks per 'a')

typedef float v2f __attribute__((ext_vector_type(2)));
typedef float v8f __attribute__((ext_vector_type(8)));

// ---------------------------------------------------------------------------
// Tensor Data Mover: 1-D contiguous f32 DMA (global -> LDS).
// Descriptor per CDNA5 ISA ch.8: group0 = {count/type/lds_addr/global_addr},
// group1 = {data_size, tensor_dim0, tile_dim0}, groups 2/3 zero (<=2D tile).
// ---------------------------------------------------------------------------
#if defined(__HIP_DEVICE_COMPILE__) && \
    __has_builtin(__builtin_amdgcn_tensor_load_to_lds) && \
    __has_builtin(__builtin_amdgcn_s_wait_tensorcnt)
#define HAVE_TDM 1
#else
#define HAVE_TDM 0
#endif

#if HAVE_TDM
typedef unsigned int v4u __attribute__((ext_vector_type(4)));
typedef int v4i __attribute__((ext_vector_type(4)));
typedef int v8i __attribute__((ext_vector_type(8)));

__device__ __forceinline__ void tdm_load_1d_f32(const float* gsrc,
                                                unsigned lds_byte_off,
                                                unsigned nelem) {
  const unsigned long long ga = (unsigned long long)gsrc;
  v4u g0;
  g0.x = 1u;                                              // count=1, user mode
  g0.y = lds_byte_off;                                    // lds_addr (bytes)
  g0.z = (unsigned)(ga & 0xffffffffull);                  // global_addr[31:0]
  g0.w = (unsigned)((ga >> 32) & 0x1ffffffull)            // global_addr[56:32]
         | (2u << 30);                                    // type = 2 (image)
  v8i g1;
  g1[0] = (int)(2u << 16);                 // wg_mask=0, data_size=2 (4 bytes)
  g1[1] = (int)((nelem & 0xffffu) << 16);  // tensor_dim0[15:0]  -> bits 63:48
  g1[2] = (int)(nelem >> 16);              // tensor_dim0[31:16] -> bits 79:64
  g1[3] = (int)((nelem & 0xffffu) << 16);  // tile_dim0          -> bits 127:112
  g1[4] = 0; g1[5] = 0; g1[6] = 0; g1[7] = 0;
  v4i gz = {0, 0, 0, 0};
#if defined(__clang_major__) && (__clang_major__ >= 23)
  v8i gz8 = {0, 0, 0, 0, 0, 0, 0, 0};
  __builtin_amdgcn_tensor_load_to_lds(g0, g1, gz, gz, gz8, 0);
#else
  __builtin_amdgcn_tensor_load_to_lds(g0, g1, gz, gz, 0);
#endif
  __builtin_amdgcn_s_wait_tensorcnt(0);
}
#endif

// ---------------------------------------------------------------------------
// One 16x16 tile of C = A(128x128) @ B(128x128), fp32 WMMA 16x16x4.
// Fragment layouts per CDNA5 ISA 7.12.2.
// ---------------------------------------------------------------------------
__device__ __forceinline__ void wmma_tile_128(const float* __restrict__ A,
                                              const float* __restrict__ B,
                                              float* __restrict__ C,
                                              int tm, int tn, int lane) {
  const int half = lane >> 4;
  const int l15  = lane & 15;
  const int m    = tm * 16 + l15;
#if __has_builtin(__builtin_amdgcn_wmma_f32_16x16x4_f32)
  v8f c = {};
  for (int k0 = 0; k0 < NBF; k0 += 4) {
    const int ka = k0 + 2 * half;
    v2f a, b;
    a.x = A[m * NBF + ka];
    a.y = A[m * NBF + ka + 1];
    b.x = B[ka * NBF + tn * 16 + l15];
    b.y = B[(ka + 1) * NBF + tn * 16 + l15];
    c = __builtin_amdgcn_wmma_f32_16x16x4_f32(false, a, false, b,
                                              (short)0, c, false, false);
  }
  for (int j = 0; j < 8; ++j)
    C[(tm * 16 + j + 8 * half) * NBF + tn * 16 + l15] = c[j];
#else
  for (int j = 0; j < 8; ++j) {
    const int row = tm * 16 + j + 8 * half;
    float acc = 0.0f;
    for (int k = 0; k < NBF; ++k)
      acc += A[row * NBF + k] * B[k * NBF + tn * 16 + l15];
    C[row * NBF + tn * 16 + l15] = acc;
  }
#endif
}

// C = A @ B, 128x128x128 fp32.  grid=8 blocks x 256 threads; wave -> one tile.
__global__ void gemm128_kernel(const float* __restrict__ A,
                               const float* __restrict__ B,
                               float* __restrict__ C) {
  const int tile = blockIdx.x * 8 + (threadIdx.x >> 5);  // 0..63
  wmma_tile_128(A, B, C, tile >> 3, tile & 7, threadIdx.x & 31);
}

// ---------------------------------------------------------------------------
// Fused J/K streaming kernel: one pass over G (1 GB / iteration, HBM-bound).
//   J[a,b] += G[a,b,c,d] * D[c,d]    K[a,c] += G[a,b,c,d] * D[b,d]
// Grid = 128 a  x  4 c-blocks = 512 workgroups.  D resident in LDS via TDM.
// J merged with global float atomics (zeroed each iteration); K exclusive.
// ---------------------------------------------------------------------------
__global__ void jk_kernel(const float* __restrict__ G,
                          const float* __restrict__ D,
                          float* __restrict__ J,
                          float* __restrict__ K) {
  extern __shared__ float lds[];
  float* Dl = lds;             // 16384 : D copy (byte offset 0 for TDM)
  float* Kl = Dl + NBF2;       // CBLK  : K[a,c0:c0+CBLK] accumulators
  float* Jp = Kl + CBLK;       // 256   : per-thread J partials

  const int tid = threadIdx.x;             // 256 threads
  const int a   = blockIdx.x >> 2;         // 0..127
  const int c0  = (blockIdx.x & 3) * CBLK; // 0,32,64,96

#if HAVE_TDM
  if (tid < 32) tdm_load_1d_f32(D, 0u, (unsigned)NBF2);  // one DMA, wave 0
#else
  for (int i = tid; i < NBF2; i += 256) Dl[i] = D[i];
#endif
  if (tid < CBLK) Kl[tid] = 0.0f;
  __syncthreads();

  const int b  = tid >> 1;           // each b owned by 2 threads
  const int dh = (tid & 1) * 64;     // d half-range
  const float* Drb = Dl + b * NBF + dh;

  float regJ = 0.0f;
  for (int ci = 0; ci < CBLK; ++ci) {
    const int c = c0 + ci;
    const float* g   = G + ((((size_t)a * NBF + b) * NBF + c) * NBF + dh);
    const float* Drc = Dl + c * NBF + dh;
    __builtin_prefetch(g + NBF, 0, 0);     // next c-row -> global_prefetch_b8
    float jp = 0.0f, kp = 0.0f;
    for (int d = 0; d < 64; d += 4) {
      float4 g4 = *reinterpret_cast<const float4*>(g + d);
      float4 dc = *reinterpret_cast<const float4*>(Drc + d);  // ds_load_b128
      float4 db = *reinterpret_cast<const float4*>(Drb + d);  // ds_load_b128
      jp += g4.x * dc.x + g4.y * dc.y + g4.z * dc.z + g4.w * dc.w;
      kp += g4.x * db.x + g4.y * db.y + g4.z * db.z + g4.w * db.w;
    }
    regJ += jp;
    for (int off = 16; off > 0; off >>= 1) kp += __shfl_xor(kp, off, 32);
    if ((tid & 31) == 0) atomicAdd(&Kl[ci], kp);   // ds_add_f32
  }

  Jp[tid] = regJ;
  __syncthreads();
  if (tid < NBF)
    atomicAdd(&J[a * NBF + tid], Jp[2 * tid] + Jp[2 * tid + 1]);  // global f32 atomic
  if (tid < CBLK)
    K[a * NBF + c0 + tid] = Kl[tid];
}

__global__ void buildF_kernel(const float* __restrict__ H,
                              const float* __restrict__ J,
                              const float* __restrict__ K,
                              float* __restrict__ F) {
  const int i = blockIdx.x * blockDim.x + threadIdx.x;
  if (i < NBF2) F[i] = H[i] + 2.0f * J[i] - K[i];
}

__global__ void zero_kernel(float* __restrict__ p, int n) {
  const int i = blockIdx.x * blockDim.x + threadIdx.x;
  if (i < n) p[i] = 0.0f;
}

// ---------------------------------------------------------------------------
// Canonical (Palser-Manolopoulos) purification: P = rank-16 spectral
// projector onto lowest eigenvectors of Fp.  Single workgroup; D, D2, D3
// live in LDS (3 x 64 KB < 320 KB/WGP); inner GEMMs are fp32 WMMA.
// ---------------------------------------------------------------------------
__global__ void purify_kernel(const float* __restrict__ Fp,
                              float* __restrict__ P) {
  extern __shared__ float lds[];
  float* Dm  = lds;               // 16384
  float* D2  = Dm + NBF2;         // 16384 (byte offset 65536 for TDM)
  float* D3  = D2 + NBF2;         // 16384
  float* red = D3 + NBF2;         // 256 reduction scratch
  float* sc  = red + 256;         // broadcast scalars

  const int tid  = threadIdx.x;   // 256 = 8 waves
  const int lane = tid & 31;
  const int wave = tid >> 5;

#if HAVE_TDM
  if (tid < 32) tdm_load_1d_f32(Fp, (unsigned)(NBF2 * sizeof(float)),
                                (unsigned)NBF2);         // stage Fp in D2
#else
  for (int i = tid; i < NBF2; i += 256) D2[i] = Fp[i];
#endif
  __syncthreads();

  // Gershgorin bounds + trace of Fp (row per thread, tid < 128)
  float diag = 0.0f, off = 0.0f;
  if (tid < NBF) {
    for (int j = 0; j < NBF; ++j) {
      const float v = D2[tid * NBF + j];
      if (j == tid) diag = v; else off += fabsf(v);
    }
  }
  red[tid] = (tid < NBF) ? diag : 0.0f;
  __syncthreads();
  for (int s = 128; s > 0; s >>= 1) { if (tid < s) red[tid] += red[tid + s]; __syncthreads(); }
  if (tid == 0) sc[0] = red[0];
  __syncthreads();
  red[tid] = (tid < NBF) ? (diag + off) : -3.4e38f;
  __syncthreads();
  for (int s = 128; s > 0; s >>= 1) { if (tid < s) red[tid] = fmaxf(red[tid], red[tid + s]); __syncthreads(); }
  if (tid == 0) sc[1] = red[0];
  __syncthreads();
  red[tid] = (tid < NBF) ? (diag - off) : 3.4e38f;
  __syncthreads();
  for (int s = 128; s > 0; s >>= 1) { if (tid < s) red[tid] = fminf(red[tid], red[tid + s]); __syncthreads(); }
  if (tid == 0) sc[2] = red[0];
  __syncthreads();

  const float mu  = sc[0] / (float)NBF;
  const float hi  = sc[1], lo = sc[2];
  const float lam = fminf(NDOCC_F / (hi - mu + 1e-12f),
                          ((float)NBF - NDOCC_F) / (mu - lo + 1e-12f));

  // D0 = (lam/n)(mu*I - Fp) + (N/n) I
  for (int i = tid; i < NBF2; i += 256) {
    const int r = i >> 7, c = i & (NBF - 1);
    float v = -(lam / (float)NBF) * D2[i];
    if (r == c) v += (lam / (float)NBF) * mu + NDOCC_F / (float)NBF;
    Dm[i] = v;
  }

  for (int it = 0; it < N_PUR; ++it) {
    __syncthreads();
    for (int t = wave; t < 64; t += 8)               // D2 = Dm @ Dm
      wmma_tile_128(Dm, Dm, D2, t >> 3, t & 7, lane);
    __syncthreads();
    for (int t = wave; t < 64; t += 8)               // D3 = D2 @ Dm
      wmma_tile_128(D2, Dm, D3, t >> 3, t & 7, lane);
    __syncthreads();

    float dd = 0.0f, d2v = 0.0f, d3v = 0.0f;
    if (tid < NBF) {
      const int di = tid * NBF + tid;
      dd = Dm[di]; d2v = D2[di]; d3v = D3[di];
    }
    red[tid] = dd - d2v;
    __syncthreads();
    for (int s = 128; s > 0; s >>= 1) { if (tid < s) red[tid] += red[tid + s]; __syncthreads(); }
    if (tid == 0) sc[0] = red[0];
    __syncthreads();
    red[tid] = d2v - d3v;
    __syncthreads();
    for (int s = 128; s > 0; s >>= 1) { if (tid < s) red[tid] += red[tid + s]; __syncthreads(); }
    if (tid == 0) sc[1] = red[0];
    __syncthreads();

    const float t1s  = sc[0], t2s = sc[1];
    const bool  conv = fabsf(t1s) < 1e-10f;
    float cn = t2s / (t1s + ((t1s >= 0.0f) ? 1e-30f : -1e-30f));
    cn = fminf(fmaxf(cn, 1e-6f), 1.0f - 1e-6f);

    for (int i = tid; i < NBF2; i += 256) {
      const float d = Dm[i], a2 = D2[i], a3 = D3[i];
      float nd;
      if (conv)            nd = d;
      else if (cn >= 0.5f) nd = ((1.0f + cn) * a2 - a3) / cn;
      else                 nd = ((1.0f - 2.0f * cn) * d + (1.0f + cn) * a2 - a3) / (1.0f - cn);
      Dm[i] = nd;
    }
  }
  __syncthreads();
  for (int i = tid; i < NBF2; i += 256) P[i] = Dm[i];
}

// E = sum((F+H) .* D_new) + Enuc
__global__ void energy_kernel(const float* __restrict__ F,
                              const float* __restrict__ H,
                              const float* __restrict__ Dn,
                              const float* __restrict__ Enuc,
                              float* __restrict__ out) {
  __shared__ float red[256];
  float acc = 0.0f;
  for (int i = threadIdx.x; i < NBF2; i += 256) acc += (F[i] + H[i]) * Dn[i];
  red[threadIdx.x] = acc;
  __syncthreads();
  for (int s = 128; s > 0; s >>= 1) {
    if (threadIdx.x < s) red[threadIdx.x] += red[threadIdx.x + s];
    __syncthreads();
  }
  if (threadIdx.x == 0) out[0] = red[0] + Enuc[0];
}

extern "C" void kernel_launch(void* const* d_in, const int* in_sizes, int n_in,
                              void* d_out, int out_size, void* d_ws, size_t ws_size,
                              hipStream_t stream) {
  (void)in_sizes; (void)n_in; (void)out_size; (void)ws_size;
  const float* H    = (const float*)d_in[0];
  const float* A    = (const float*)d_in[1];
  const float* G    = (const float*)d_in[2];
  const float* Enuc = (const float*)d_in[3];
  // d_in[4] = ndocc (int) — fixed at 16, baked into kernels.
  float* out = (float*)d_out;
  float* ws  = (float*)d_ws;

  float* J  = ws;
  float* K  = ws + 1 * NBF2;
  float* F  = ws + 2 * NBF2;
  float* Fp = ws + 3 * NBF2;
  float* T  = ws + 4 * NBF2;
  float* P  = ws + 5 * NBF2;
  float* Da = ws + 6 * NBF2;
  float* Db = ws + 7 * NBF2;

  const size_t jk_lds  = (size_t)(NBF2 + CBLK + 256) * sizeof(float);
  const size_t pur_lds = (size_t)(3 * NBF2 + 256 + 16) * sizeof(float);

  zero_kernel<<<64, 256, 0, stream>>>(Da, NBF2);   // D0 = 0

  float* Dcur = Da;
  float* Dnew = Db;
  for (int it = 0; it < N_ITER; ++it) {
    zero_kernel<<<64, 256, 0, stream>>>(J, NBF2);                // J accumulators
    jk_kernel<<<512, 256, jk_lds, stream>>>(G, Dcur, J, K);      // one pass over G
    buildF_kernel<<<64, 256, 0, stream>>>(H, J, K, F);           // F = H + 2J - K
    gemm128_kernel<<<8, 256, 0, stream>>>(F, A, T);              // T  = F @ A
    gemm128_kernel<<<8, 256, 0, stream>>>(A, T, Fp);             // Fp = A @ F @ A
    purify_kernel<<<1, 256, pur_lds, stream>>>(Fp, P);           // P: rank-16 projector
    gemm128_kernel<<<8, 256, 0, stream>>>(P, A, T);              // T     = P @ A
    gemm128_kernel<<<8, 256, 0, stream>>>(A, T, Dnew);           // D_new = A @ P @ A
    energy_kernel<<<1, 256, 0, stream>>>(F, H, Dnew, Enuc, out); // last write wins
    float* tmp = Dcur; Dcur = Dnew; Dnew = tmp;
  }
}